// YuGCN_16277926052608
// MI455X (gfx1250) — compile-verified
//
#include <hip/hip_runtime.h>
#include <stdint.h>
#include <stddef.h>

// Problem constants (match reference).
#define BB 64
#define NN 512
#define TT 50
#define FF 32
#define EE 32768
#define MT 8   // M-tiles (16 rows each) per workgroup in cheb_gemm

typedef __attribute__((ext_vector_type(2))) float v2f;
typedef __attribute__((ext_vector_type(4))) float v4f;
typedef __attribute__((ext_vector_type(8))) float v8f;

// ---------------------------------------------------------------------------
// Graph preprocessing (deterministic, no float atomics).
// ---------------------------------------------------------------------------

// Per destination node: weighted degree of sources (self-loops masked) and
// incoming-edge count. 512 node-threads scan all E edges; ~17M scalar ops,
// negligible and fully deterministic.
__global__ void build_deg_cnt(const int* __restrict__ ei, const float* __restrict__ ew,
                              float* __restrict__ deg, int* __restrict__ cnt) {
  int n = blockIdx.x * blockDim.x + threadIdx.x;
  if (n >= NN) return;
  float d = 0.f;
  int c = 0;
  for (int e = 0; e < EE; ++e) {
    int s = ei[e];
    int t = ei[EE + e];
    if (s == n && s != t) d += ew[e];
    if (t == n) ++c;
  }
  deg[n] = d;
  cnt[n] = c;
}

// Single-block exclusive prefix scan over the 512 per-node counts -> rowptr.
__global__ void scan_kernel(const int* __restrict__ cnt, int* __restrict__ rowptr) {
  __shared__ int tmp[NN];
  int t = threadIdx.x;
  tmp[t] = cnt[t];
  __syncthreads();
  for (int off = 1; off < NN; off <<= 1) {
    int v = (t >= off) ? tmp[t - off] : 0;
    __syncthreads();
    tmp[t] += v;
    __syncthreads();
  }
  rowptr[t + 1] = tmp[t];
  if (t == 0) rowptr[0] = 0;
}

// CSR fill, ordered by edge id (deterministic). norm = -(dinv[s]*w*dinv[d]).
__global__ void build_csr(const int* __restrict__ ei, const float* __restrict__ ew,
                          const float* __restrict__ deg, const int* __restrict__ rowptr,
                          int* __restrict__ psrc, float* __restrict__ pnorm) {
  int n = blockIdx.x * blockDim.x + threadIdx.x;
  if (n >= NN) return;
  int pos = rowptr[n];
  float dd = deg[n];
  float dinv_d = dd > 0.f ? rsqrtf(dd) : 0.f;
  for (int e = 0; e < EE; ++e) {
    int t = ei[EE + e];
    if (t != n) continue;
    int s = ei[e];
    float wv = (s == t) ? 0.f : ew[e];
    float ds = deg[s];
    float dinv_s = ds > 0.f ? rsqrtf(ds) : 0.f;
    psrc[pos] = s;
    pnorm[pos] = -(dinv_s * wv * dinv_d);
    ++pos;
  }
}

// ---------------------------------------------------------------------------
// Sparse propagation: z[b,n,:] = sum_{e: dst==n} norm[e] * h[b, src[e], :]
// One block per (b,n); threads cover the channel dim. Pure gather, no atomics.
// ---------------------------------------------------------------------------
__global__ void prop_kernel(const float* __restrict__ h, const float* __restrict__ pnorm,
                            const int* __restrict__ psrc, const int* __restrict__ rowptr,
                            float* __restrict__ z, int C) {
  int bn = blockIdx.x;          // 0 .. B*N-1
  int b = bn >> 9;              // / 512
  int n = bn & (NN - 1);
  int c = threadIdx.x;
  if (c >= C) return;
  const float* hb = h + (size_t)b * NN * C;
  float s = 0.f;
  int e1 = rowptr[n + 1];
  for (int i = rowptr[n]; i < e1; ++i)
    s = fmaf(pnorm[i], hb[psrc[i] * C + c], s);
  z[(size_t)bn * C + c] = s;
}

// ---------------------------------------------------------------------------
// ChebConv K=2 GEMM via V_WMMA_F32_16X16X4_F32 (full fp32 precision).
// out[m, 0:32] = [H | Z][m, 0:2C] @ W[0:2C, 0:32] + bias, optional ReLU.
//
// One wave per workgroup; the wave stages W ([2C,32] f32, <= 12.8 KB) into LDS
// once — via the Tensor Data Mover when available (tensor_load_to_lds +
// s_wait_tensorcnt), else a cooperative b128 copy — then sweeps MT=8 16-row
// M-tiles reading A as 8-byte pairs from global and B fragments from LDS.
// EXEC is all-ones at every WMMA (no divergent control flow in the wave).
//
// f32 WMMA fragment layouts (ISA 7.12.2):
//   A 16x4:  lanes 0-15 -> M=lane (K-pair 0/1), lanes 16-31 -> M=lane-16
//            (K-pair 2/3); VGPR0 = even K of pair base, VGPR1 = +1.
//   B 4x16:  lanes 0-15 -> N=lane (K 0/1), lanes 16-31 -> N=lane-16 (K 2/3).
//   C/D 16x16: VGPR r -> row r (lanes 0-15) / row r+8 (lanes 16-31), N=lane%16.
// ---------------------------------------------------------------------------
__global__ __launch_bounds__(32)
void cheb_gemm(const float* __restrict__ H, const float* __restrict__ Z,
               const float* __restrict__ W, const float* __restrict__ bias,
               float* __restrict__ out, int C, int relu) {
  __shared__ float Wlds[2 * TT * FF];      // max 100*32 floats = 12.8 KB
  const int lane = threadIdx.x;            // 0..31
  const int K = 2 * C;                     // 100 (layer1) or 64; both %4 == 0

#if defined(__gfx1250__) && __has_builtin(__builtin_amdgcn_tensor_load_to_lds) && \
    __has_builtin(__builtin_amdgcn_s_wait_tensorcnt)
  {
    // Tensor DMA descriptor (CDNA5 ISA):
    // 2-D f32 tensor [K rows x 32 cols], row stride 32, whole tensor as tile,
    // destination LDS offset 0 (Wlds is the kernel's only LDS allocation).
    typedef __attribute__((ext_vector_type(4))) unsigned int v4u;
    typedef __attribute__((ext_vector_type(8))) int v8i;
    typedef __attribute__((ext_vector_type(4))) int v4i;
    unsigned long long ga = (unsigned long long)(uintptr_t)W;
    v4u g0;
    g0[0] = 1u;                                          // count=1, user mode
    g0[1] = 0u;                                          // lds_addr = 0
    g0[2] = (unsigned int)(ga & 0xffffffffu);            // global_addr[31:0]
    g0[3] = (unsigned int)((ga >> 32) & 0x01ffffffu)     // global_addr[56:32]
            | (2u << 30);                                // type = 2 (image)
    v8i g1;
    g1[0] = (int)(2u << 16);                             // data_size = 4 bytes
    g1[1] = (int)((unsigned)FF << 16);                   // tensor_dim0 = 32
    g1[2] = (int)((unsigned)K << 16);                    // tensor_dim1 = K
    g1[3] = (int)((unsigned)FF << 16);                   // tile_dim0 = 32
    g1[4] = (int)(unsigned)K;                            // tile_dim1 = K
    g1[5] = (int)FF;                                     // tensor_dim0_stride = 32
    g1[6] = 0;
    g1[7] = 0;
    v4i gz4;
    gz4[0] = 0; gz4[1] = 0; gz4[2] = 0; gz4[3] = 0;      // groups 2/3 unused (2-D)
    v8i gz8;
    gz8[0] = 0; gz8[1] = 0; gz8[2] = 0; gz8[3] = 0;
    gz8[4] = 0; gz8[5] = 0; gz8[6] = 0; gz8[7] = 0;
    // 6-arg form (amdgpu-toolchain / clang-23 signature).
    __builtin_amdgcn_tensor_load_to_lds(g0, g1, gz4, gz4, gz8, 0);
    __builtin_amdgcn_s_wait_tensorcnt(0);
    // The DMA's LDS writes are invisible to the compiler (destination lives
    // inside the descriptor). Escape Wlds into an asm memory clobber so the
    // subsequent ds_load reads are not folded to undef / the LDS allocation
    // is not eliminated.
    float* wp = &Wlds[0];
    asm volatile("" : "+r"(wp) : : "memory");
  }
#else
  for (int i = lane * 4; i < K * FF; i += 32 * 4)
    *(v4f*)&Wlds[i] = *(const v4f*)&W[i];
  __syncthreads();
#endif

  const int kg = (lane >> 4) * 2;          // K-pair select within fragment
  const int nb = lane & 15;

  for (int t = 0; t < MT; ++t) {
    const int m0 = (blockIdx.x * MT + t) * 16;
    const int mr = m0 + (lane & 15);
    const float* __restrict__ Hrow = H + (size_t)mr * C;
    const float* __restrict__ Zrow = Z + (size_t)mr * C;
    v8f acc0 = {};                         // output cols 0..15
    v8f acc1 = {};                         // output cols 16..31
    for (int k0 = 0; k0 < K; k0 += 4) {
      const int ka = k0 + kg;
      // K-pairs start at even k and C is even, so a pair never straddles the
      // H|Z boundary -> single 8-byte load from whichever source holds it.
      v2f a = (ka < C) ? *(const v2f*)(Hrow + ka) : *(const v2f*)(Zrow + (ka - C));
      v2f b0, b1;
      b0[0] = Wlds[ka * FF + nb];
      b0[1] = Wlds[(ka + 1) * FF + nb];
      b1[0] = Wlds[ka * FF + 16 + nb];
      b1[1] = Wlds[(ka + 1) * FF + 16 + nb];
      acc0 = __builtin_amdgcn_wmma_f32_16x16x4_f32(false, a, false, b0, (short)0, acc0, false, false);
      acc1 = __builtin_amdgcn_wmma_f32_16x16x4_f32(false, a, false, b1, (short)0, acc1, false, false);
    }
    const int rb = (lane >> 4) * 8;
    for (int r = 0; r < 8; ++r) {
      const int row = m0 + rb + r;
      float v0 = acc0[r] + bias[nb];
      float v1 = acc1[r] + bias[16 + nb];
      if (relu) { v0 = fmaxf(v0, 0.f); v1 = fmaxf(v1, 0.f); }
      out[row * FF + nb] = v0;
      out[row * FF + 16 + nb] = v1;
    }
  }
}

// ---------------------------------------------------------------------------
// Generic dense GEMM tile via f32 WMMA: out[M,N] = A[M,K] @ W[K,N] + bias.
// Grid: (N/16, M/16); one wave per 16x16 tile. K must be a multiple of 4.
// A fragment pairs are contiguous -> b64 loads; prefetch hides the K=16384
// stream for FC1 (global_prefetch_b8).
// ---------------------------------------------------------------------------
__global__ __launch_bounds__(32)
void fc_gemm(const float* __restrict__ A, const float* __restrict__ W,
             const float* __restrict__ bias, float* __restrict__ out,
             int N, int K) {
  const int lane = threadIdx.x;
  const int n0 = blockIdx.x * 16;
  const int m0 = blockIdx.y * 16;
  const int mr = m0 + (lane & 15);
  const int nb = n0 + (lane & 15);
  const int kg = (lane >> 4) * 2;
  const float* __restrict__ Arow = A + (size_t)mr * K;
  v8f acc = {};
  for (int k0 = 0; k0 < K; k0 += 4) {
    if (k0 + 256 < K)
      __builtin_prefetch(Arow + k0 + 256, 0, 0);
    v2f a = *(const v2f*)(Arow + k0 + kg);
    v2f b;
    b[0] = W[(k0 + kg) * N + nb];
    b[1] = W[(k0 + kg + 1) * N + nb];
    acc = __builtin_amdgcn_wmma_f32_16x16x4_f32(false, a, false, b, (short)0, acc, false, false);
  }
  const int rb = (lane >> 4) * 8;
  const int col = n0 + (lane & 15);
  for (int r = 0; r < 8; ++r)
    out[(m0 + rb + r) * N + col] = acc[r] + bias[col];
}

// Final tiny layer: [64,128] @ [128,2] + bias -> d_out. N=2 < 16, scalar is fine.
__global__ void fc3_kernel(const float* __restrict__ A, const float* __restrict__ W,
                           const float* __restrict__ bias, float* __restrict__ out) {
  int idx = blockIdx.x * blockDim.x + threadIdx.x;
  if (idx >= BB * 2) return;
  int m = idx >> 1, n = idx & 1;
  float s = bias[n];
  for (int k = 0; k < 128; ++k)
    s = fmaf(A[m * 128 + k], W[k * 2 + n], s);
  out[idx] = s;
}

// ---------------------------------------------------------------------------
// Launch
// ---------------------------------------------------------------------------
extern "C" void kernel_launch(void* const* d_in, const int* in_sizes, int n_in,
                              void* d_out, int out_size, void* d_ws, size_t ws_size,
                              hipStream_t stream) {
  (void)in_sizes; (void)n_in; (void)out_size; (void)ws_size;

  const float* x  = (const float*)d_in[0];        // [B, N, T]
  const int*   ei = (const int*)d_in[1];          // [2, E]
  const float* ew = (const float*)d_in[2];        // [E]
  const float* cW[6] = { (const float*)d_in[3],  (const float*)d_in[5],
                         (const float*)d_in[7],  (const float*)d_in[9],
                         (const float*)d_in[11], (const float*)d_in[13] };
  const float* cb[6] = { (const float*)d_in[4],  (const float*)d_in[6],
                         (const float*)d_in[8],  (const float*)d_in[10],
                         (const float*)d_in[12], (const float*)d_in[14] };
  const float* fc1W = (const float*)d_in[15];
  const float* fc1b = (const float*)d_in[16];
  const float* fc2W = (const float*)d_in[17];
  const float* fc2b = (const float*)d_in[18];
  const float* fc3W = (const float*)d_in[19];
  const float* fc3b = (const float*)d_in[20];

  // Workspace carve-up (all regions fully overwritten every call).
  char* ws = (char*)d_ws;
  size_t off = 0;
  auto alloc = [&](size_t bytes) -> void* {
    void* p = (void*)(ws + off);
    off += (bytes + 255) & ~(size_t)255;
    return p;
  };
  float* deg    = (float*)alloc(NN * sizeof(float));
  int*   cnt    = (int*)  alloc(NN * sizeof(int));
  int*   rowptr = (int*)  alloc((NN + 1) * sizeof(int));
  int*   psrc   = (int*)  alloc(EE * sizeof(int));
  float* pnorm  = (float*)alloc(EE * sizeof(float));
  float* z      = (float*)alloc((size_t)BB * NN * TT * sizeof(float)); // max C = 50
  float* bufA   = (float*)alloc((size_t)BB * NN * FF * sizeof(float));
  float* bufB   = (float*)alloc((size_t)BB * NN * FF * sizeof(float));
  float* fco1   = (float*)alloc((size_t)BB * 256 * sizeof(float));
  float* fco2   = (float*)alloc((size_t)BB * 128 * sizeof(float));

  // Graph preprocessing (shared across batch and all 6 layers).
  build_deg_cnt<<<2, 256, 0, stream>>>(ei, ew, deg, cnt);
  scan_kernel<<<1, NN, 0, stream>>>(cnt, rowptr);
  build_csr<<<2, 256, 0, stream>>>(ei, ew, deg, rowptr, psrc, pnorm);

  const int M = BB * NN;                     // 32768 rows

  // conv1: C = T = 50, ReLU.
  prop_kernel<<<M, 64, 0, stream>>>(x, pnorm, psrc, rowptr, z, TT);
  cheb_gemm<<<M / (16 * MT), 32, 0, stream>>>(x, z, cW[0], cb[0], bufA, TT, 1);

  // conv2..conv6: C = 32. ReLU on conv2..conv5, none on conv6.
  float* hin = bufA;
  float* hout = bufB;
  for (int L = 1; L < 6; ++L) {
    prop_kernel<<<M, 32, 0, stream>>>(hin, pnorm, psrc, rowptr, z, FF);
    cheb_gemm<<<M / (16 * MT), 32, 0, stream>>>(hin, z, cW[L], cb[L], hout, FF, (L < 5) ? 1 : 0);
    float* t = hin; hin = hout; hout = t;
  }

  // FC head: hin is [B, N*F] = [64, 16384] flat (matches reshape semantics).
  fc_gemm<<<dim3(256 / 16, BB / 16), 32, 0, stream>>>(hin,  fc1W, fc1b, fco1, 256, NN * FF);
  fc_gemm<<<dim3(128 / 16, BB / 16), 32, 0, stream>>>(fco1, fc2W, fc2b, fco2, 128, 256);
  fc3_kernel<<<1, 128, 0, stream>>>(fco2, fc3W, fc3b, (float*)d_out);
}